// KipfAndWillingConv_74569222193317
// MI455X (gfx1250) — compile-verified
//
#include <hip/hip_runtime.h>
#include <hip/hip_bf16.h>

typedef float v2f __attribute__((ext_vector_type(2)));
typedef float v8f __attribute__((ext_vector_type(8)));

#define KDIM 256   // N_FEAT
#define NDIM 256   // N_FILT

// ---------------------------------------------------------------------------
// Zero the output (it is accumulated via atomics, and the harness poisons it).
// RT stores intentionally: pulls `out` lines into L2 dirty, where the edge
// phase's f32 atomics will then resolve.
// ---------------------------------------------------------------------------
__global__ void zero_f32(float* __restrict__ p, long long n4) {
    long long i      = (long long)blockIdx.x * blockDim.x + threadIdx.x;
    long long stride = (long long)gridDim.x * blockDim.x;
    float4 z = make_float4(0.f, 0.f, 0.f, 0.f);
    for (; i < n4; i += stride) ((float4*)p)[i] = z;
}

// ---------------------------------------------------------------------------
// Repack filters so each lane's B fragment (two K-values, one column) is a
// contiguous 8-byte pair -> single global_load_b64 instead of 2x b32 1KB apart.
// Layout: Fp[((k/4)*2 + (k/2)%2)*256 + n][k%2]  (v2f units)
//   lanes 0-15  (lsel=0) read pair {F[4g+0][n], F[4g+1][n]}
//   lanes 16-31 (lsel=1) read pair {F[4g+2][n], F[4g+3][n]}
// ---------------------------------------------------------------------------
__global__ __launch_bounds__(256) void pack_filters(
    const float* __restrict__ F, float* __restrict__ Fp)
{
    const int idx = blockIdx.x * 256 + threadIdx.x;   // 0 .. 65535
    const int k = idx >> 8;
    const int n = idx & 255;
    const int g = k >> 2;          // K group of 4
    const int j = (k >> 1) & 1;    // pair within group (lsel)
    const int p = k & 1;           // element within pair
    Fp[(((size_t)g * 2 + j) * NDIM + n) * 2 + p] = F[idx];
}

// ---------------------------------------------------------------------------
// XF = X @ F  via V_WMMA_F32_16X16X4_F32 (fp32 matrix core, wave32).
//
// Block = 256 threads = 8 waves. Wave w handles:
//   row tile  = blockIdx.x*4 + (w>>1)   (16 rows)
//   col half  = (w&1)*128               (8 tiles of 16 cols, A reused 8x)
//
// Per-lane fragment layout (ISA 7.12.2, 32-bit 16x4 A / 4x16 B / 16x16 C):
//   lanes 0-15 : rows/cols N=lane,      K pair {k, k+1}
//   lanes 16-31: rows/cols N=lane-16,   K pair {k+2, k+3}
//   C/D VGPR v : M = v (lanes 0-15) / M = v+8 (lanes 16-31)
// All branches are wave-uniform => EXEC all-ones for every WMMA.
// Per K-step: 1x b64 (A) + 8x b64 (B, packed) + 8 WMMA.
// ---------------------------------------------------------------------------
__global__ __launch_bounds__(256) void gemm_wmma_f32(
    const float* __restrict__ X, const float* __restrict__ Fp,
    float* __restrict__ XF, int nrows)
{
    const int lane = threadIdx.x & 31;
    const int wave = threadIdx.x >> 5;

    const int rowTile = blockIdx.x * 4 + (wave >> 1);
    const int m0 = rowTile * 16;
    if (m0 >= nrows) return;                 // uniform per wave
    const int n0 = (wave & 1) * 128;

    const int lsel = lane >> 4;              // 0 -> K{0,1}, 1 -> K{2,3}
    const int lrow = lane & 15;

    const float* arow  = X + (size_t)(m0 + lrow) * KDIM + (lsel * 2);
    const v2f*   bbase = (const v2f*)Fp + (size_t)lsel * NDIM + n0 + lrow;

    v8f acc[8] = {};

    for (int k = 0; k < KDIM; k += 4) {
        v2f a = *(const v2f*)(arow + k);             // b64: X[row][k+2*lsel..+1]
        const v2f* b0 = bbase + (size_t)(k >> 2) * (2 * NDIM);
        #pragma unroll
        for (int t = 0; t < 8; ++t) {
            v2f b = b0[t * 16];                      // b64: packed B pair
            acc[t] = __builtin_amdgcn_wmma_f32_16x16x4_f32(
                /*neg_a=*/false, a, /*neg_b=*/false, b,
                /*c_mod=*/(short)0, acc[t],
                /*reuse_a=*/false, /*reuse_b=*/false);
        }
    }

    // Store D: row = m0 + v + 8*lsel, col = n0 + t*16 + lrow
    float* crow = XF + (size_t)(m0 + lsel * 8) * NDIM + n0 + lrow;
    #pragma unroll
    for (int t = 0; t < 8; ++t) {
        #pragma unroll
        for (int v = 0; v < 8; ++v) {
            crow[(size_t)v * NDIM + t * 16] = acc[t][v];
        }
    }
}

// ---------------------------------------------------------------------------
// Edge phase: out[row[e]] += w[e] * XF[col[e]], one wave32 per edge.
// Each lane owns 8 consecutive columns (2x b128 gather + 8x f32 atomics).
// XF (102 MB) and out (102 MB) both live in the 192 MB L2 -> L2-atomic bound.
// Edge index/weight streams are read-once -> nontemporal loads so the 38 MB
// stream does not evict the L2-resident XF/out working set.
// ---------------------------------------------------------------------------
__global__ __launch_bounds__(256) void edge_scatter(
    const float* __restrict__ XF, const int* __restrict__ erow,
    const int* __restrict__ ecol, const float* __restrict__ ew,
    float* __restrict__ out, int nedges)
{
    const int lane = threadIdx.x & 31;
    const long long e = (long long)blockIdx.x * 8 + (threadIdx.x >> 5);
    if (e >= nedges) return;                 // uniform per wave

    const int   r = __builtin_nontemporal_load(erow + e);
    const int   c = __builtin_nontemporal_load(ecol + e);
    const float w = __builtin_nontemporal_load(ew + e);

    const float4* src = (const float4*)(XF + (size_t)c * NDIM) + lane * 2;
    float4 v0 = src[0];                      // RT: XF rows reused ~32x from L2
    float4 v1 = src[1];

    float* dst = out + (size_t)r * NDIM + lane * 8;
    unsafeAtomicAdd(dst + 0, w * v0.x);
    unsafeAtomicAdd(dst + 1, w * v0.y);
    unsafeAtomicAdd(dst + 2, w * v0.z);
    unsafeAtomicAdd(dst + 3, w * v0.w);
    unsafeAtomicAdd(dst + 4, w * v1.x);
    unsafeAtomicAdd(dst + 5, w * v1.y);
    unsafeAtomicAdd(dst + 6, w * v1.z);
    unsafeAtomicAdd(dst + 7, w * v1.w);
}

// ---------------------------------------------------------------------------
extern "C" void kernel_launch(void* const* d_in, const int* in_sizes, int n_in,
                              void* d_out, int out_size, void* d_ws, size_t ws_size,
                              hipStream_t stream) {
    const float* x       = (const float*)d_in[0];   // [nrows, 256]
    const float* filters = (const float*)d_in[1];   // [256, 256]
    const int*   erow    = (const int*)d_in[2];     // [nedges]
    const int*   ecol    = (const int*)d_in[3];     // [nedges]
    const float* ew      = (const float*)d_in[4];   // [nedges]
    float*       out     = (float*)d_out;           // [nrows, 256]

    const int nrows  = in_sizes[0] / KDIM;
    const int nedges = in_sizes[2];

    // Workspace partition: XF [nrows*256] f32, then packed filters [256*256] f32
    float* XF = (float*)d_ws;
    float* Fp = XF + (size_t)nrows * NDIM;

    // 1) out = 0 (atomic accumulation target; harness poisons buffers)
    const long long n4 = (long long)out_size / 4;
    zero_f32<<<2048, 256, 0, stream>>>(out, n4);

    // 2) repack filters for b64 B-fragment loads
    pack_filters<<<(KDIM * NDIM) / 256, 256, 0, stream>>>(filters, Fp);

    // 3) XF = x @ filters (fp32 WMMA)
    const int rowTiles   = (nrows + 15) / 16;
    const int gemmBlocks = (rowTiles + 3) / 4;      // 4 row tiles per 8-wave block
    gemm_wmma_f32<<<gemmBlocks, 256, 0, stream>>>(x, Fp, XF, nrows);

    // 4) COO scatter-add
    const int edgeBlocks = (nedges + 7) / 8;        // 8 edges (waves) per block
    edge_scatter<<<edgeBlocks, 256, 0, stream>>>(XF, erow, ecol, ew, out, nedges);
}